// VectorQuant_35261681500804
// MI455X (gfx1250) — compile-verified
//
#include <hip/hip_runtime.h>
#include <math.h>

typedef __attribute__((ext_vector_type(2))) float v2f;
typedef __attribute__((ext_vector_type(4))) float v4f;
typedef __attribute__((ext_vector_type(8))) float v8f;
typedef __attribute__((ext_vector_type(4))) int   v4i;

// pointer-to-int4 in global (1) / LDS (3) address spaces, for async-to-LDS
typedef __attribute__((address_space(1))) v4i* gv4i_p;
typedef __attribute__((address_space(3))) v4i* lv4i_p;

#define NS_TOT  65536      // N*S samples
#define KK      1024       // codewords
#define VV      64         // vector length
#define NTILES  (KK / 16)  // 64 codeword tiles
#define ROW_PAD 68         // LDS row stride in floats (64 + 4 pad -> conflict-free)
#define BUF_F   (32 * ROW_PAD)  // floats per double-buffer half (2 tiles)

#if __has_builtin(__builtin_amdgcn_global_load_async_to_lds_b128) && \
    __has_builtin(__builtin_amdgcn_s_wait_asynccnt)
#define USE_ASYNC_LDS 1
#else
#define USE_ASYNC_LDS 0
#endif

// ---------------------------------------------------------------------------
// Kernel 1: per-codeword squared norms + zero the histogram. 1024 threads.
// ---------------------------------------------------------------------------
__global__ void vq_prep(const float* __restrict__ emb,
                        float* __restrict__ ee,
                        unsigned* __restrict__ counts) {
    int k = blockIdx.x * blockDim.x + threadIdx.x;
    if (k < KK) {
        const float* e = emb + (size_t)k * VV;
        float s = 0.f;
#pragma unroll
        for (int v = 0; v < VV; ++v) s += e[v] * e[v];
        ee[k] = s;
        counts[k] = 0u;
    }
}

// ---------------------------------------------------------------------------
// Kernel 2: main VQ. One wave = 16 samples vs all 1024 codewords via
// v_wmma_f32_16x16x4_f32. 8 waves/block share double-buffered LDS codebook
// tiles; next stage overlaps the WMMA chains (async-to-LDS when available).
// ---------------------------------------------------------------------------
__launch_bounds__(256)
__global__ void vq_main(const float* __restrict__ x,
                        const float* __restrict__ emb,
                        const float* __restrict__ ee,
                        unsigned* __restrict__ counts,
                        float* __restrict__ out0,
                        float* __restrict__ out1,
                        float* __restrict__ out2) {
    __shared__ float ebuf[2 * BUF_F];      // double-buffered 2-tile stages

    const int tid  = threadIdx.x;
    const int lane = tid & 31;
    const int wave = tid >> 5;             // 0..7
    const int hh   = lane >> 4;            // half-wave: 0 or 1
    const int col  = lane & 15;
    const int rowBase = blockIdx.x * 128 + wave * 16;

    // This thread's two float4 staging chunks (32 rows x 16 float4 = 512).
    const int sidx0 = tid;
    const int sidx1 = tid + 256;
    const int l0 = (sidx0 >> 4) * ROW_PAD + (sidx0 & 15) * 4;  // float offset
    const int l1 = (sidx1 >> 4) * ROW_PAD + (sidx1 & 15) * 4;

    // ---- A fragments: 16 steps of 16x4 fp32 A-layout.
    // lane holds row M=col; half-wave selects K sub-pair {0,1} / {2,3}.
    v2f a[16];
    {
        const v2f* xr2 = (const v2f*)(x + (size_t)(rowBase + col) * VV + 2 * hh);
#pragma unroll
        for (int s = 0; s < 16; ++s) a[s] = xr2[2 * s];
    }

    // ---- ||x_m||^2 : per-lane partial over its 32 elements, + other half.
    float xs = 0.f;
#pragma unroll
    for (int s = 0; s < 16; ++s) xs += a[s].x * a[s].x + a[s].y * a[s].y;
    xs += __shfl_xor(xs, 16, 32);          // lanes l and l^16 hold ||x_col||^2

    // Per accumulator-slot row norm: slot r holds row M = r + 8*hh.
    float xr[8];
#pragma unroll
    for (int r = 0; r < 8; ++r) xr[r] = __shfl(xs, r + 8 * hh, 32);

    float minv[8];
    int   mink[8];
#pragma unroll
    for (int r = 0; r < 8; ++r) { minv[r] = 3.4e38f; mink[r] = 0; }

    // ---- prologue: stage tiles {0,1} into buffer 0
#if USE_ASYNC_LDS
    {
        const v4f* src = (const v4f*)emb;
        __builtin_amdgcn_global_load_async_to_lds_b128(
            (gv4i_p)(src + sidx0), (lv4i_p)&ebuf[l0], 0, 0);
        __builtin_amdgcn_global_load_async_to_lds_b128(
            (gv4i_p)(src + sidx1), (lv4i_p)&ebuf[l1], 0, 0);
        __builtin_amdgcn_s_wait_asynccnt(0);
    }
#else
    {
        const v4f* src = (const v4f*)emb;
        v4f p0 = src[sidx0];
        v4f p1 = src[sidx1];
        *(v4f*)&ebuf[l0] = p0;
        *(v4f*)&ebuf[l1] = p1;
    }
#endif
    __syncthreads();

    for (int kt = 0; kt < NTILES; kt += 2) {
        const int   cur  = (kt >> 1) & 1;
        const int   nxt  = cur ^ 1;
        const bool  haveNext = (kt + 2) < NTILES;
        const float* base = &ebuf[cur * BUF_F];

        // ---- issue next stage so it overlaps the WMMA chains
#if USE_ASYNC_LDS
        if (haveNext) {
            const v4f* src = (const v4f*)(emb + (size_t)(kt + 2) * 16 * VV);
            __builtin_amdgcn_global_load_async_to_lds_b128(
                (gv4i_p)(src + sidx0), (lv4i_p)&ebuf[nxt * BUF_F + l0], 0, 0);
            __builtin_amdgcn_global_load_async_to_lds_b128(
                (gv4i_p)(src + sidx1), (lv4i_p)&ebuf[nxt * BUF_F + l1], 0, 0);
        }
#else
        v4f t0 = {}, t1 = {};
        if (haveNext) {
            const v4f* src = (const v4f*)(emb + (size_t)(kt + 2) * 16 * VV);
            t0 = src[sidx0];
            t1 = src[sidx1];
        }
#endif

        // ---- two independent 16-step WMMA accumulation chains
        v8f acc0 = {};
        v8f acc1 = {};
#pragma unroll
        for (int s = 0; s < 16; ++s) {
            v2f b0 = *(const v2f*)&base[col * ROW_PAD + 4 * s + 2 * hh];
            v2f b1 = *(const v2f*)&base[(16 + col) * ROW_PAD + 4 * s + 2 * hh];
            acc0 = __builtin_amdgcn_wmma_f32_16x16x4_f32(
                false, a[s], false, b0, (short)0, acc0, false, false);
            acc1 = __builtin_amdgcn_wmma_f32_16x16x4_f32(
                false, a[s], false, b1, (short)0, acc1, false, false);
        }

        const int   k0  = kt * 16 + col;
        const int   k1  = k0 + 16;
        const float ee0 = ee[k0];
        const float ee1 = ee[k1];
#pragma unroll
        for (int r = 0; r < 8; ++r) {
            float d0 = xr[r] - 2.f * acc0[r] + ee0;
            float d1 = xr[r] - 2.f * acc1[r] + ee1;
            if (d0 < minv[r]) { minv[r] = d0; mink[r] = k0; }
            if (d1 < minv[r]) { minv[r] = d1; mink[r] = k1; }
        }

        // ---- retire the stage and publish it
#if USE_ASYNC_LDS
        __builtin_amdgcn_s_wait_asynccnt(0);
#else
        if (haveNext) {
            *(v4f*)&ebuf[nxt * BUF_F + l0] = t0;
            *(v4f*)&ebuf[nxt * BUF_F + l1] = t1;
        }
#endif
        __syncthreads();
    }

    // ---- min+argmin across the 16 lane-columns of each half (lowest index wins ties)
#pragma unroll
    for (int r = 0; r < 8; ++r) {
        float v  = minv[r];
        int   kb = mink[r];
#pragma unroll
        for (int m = 8; m >= 1; m >>= 1) {
            float ov = __shfl_xor(v, m, 16);
            int   ok = __shfl_xor(kb, m, 16);
            if (ov < v || (ov == v && ok < kb)) { v = ov; kb = ok; }
        }
        minv[r] = v;
        mink[r] = kb;
    }

    // ---- outputs: half-wave hh owns rows r + 8*hh; 16 lanes gather 4 floats each
#pragma unroll
    for (int r = 0; r < 8; ++r) {
        const int m  = r + 8 * hh;
        const int ns = rowBase + m;
        const int kb = mink[r];

        v4f ev = ((const v4f*)(emb + (size_t)kb * VV))[col];
        v4f xv = ((const v4f*)(x + (size_t)ns * VV))[col];
        __builtin_nontemporal_store(ev, (v4f*)(out0 + (size_t)ns * VV + col * 4));

        v4f  dd = xv - ev;
        float ps = dd.x * dd.x + dd.y * dd.y + dd.z * dd.z + dd.w * dd.w;
#pragma unroll
        for (int mm = 8; mm >= 1; mm >>= 1) ps += __shfl_xor(ps, mm, 16);

        if (col == 0) {
            __builtin_nontemporal_store(ps, out1 + ns);   // exact sum((x-e)^2)
            __builtin_nontemporal_store(ps, out2 + ns);
            atomicAdd(&counts[kb], 1u);                   // deterministic histogram
        }
    }
}

// ---------------------------------------------------------------------------
// Kernel 3: entropy of the histogram. One wave.
// ---------------------------------------------------------------------------
__global__ void vq_entropy(const unsigned* __restrict__ counts,
                           float* __restrict__ ent) {
    float acc = 0.f;
    for (int k = threadIdx.x; k < KK; k += 32) {
        unsigned c = counts[k];
        if (c) {
            float p = (float)c * (1.0f / (float)NS_TOT);
            acc += p * logf(p);
        }
    }
#pragma unroll
    for (int m = 16; m >= 1; m >>= 1) acc += __shfl_xor(acc, m, 32);
    if (threadIdx.x == 0) ent[0] = -acc;
}

// ---------------------------------------------------------------------------
extern "C" void kernel_launch(void* const* d_in, const int* in_sizes, int n_in,
                              void* d_out, int out_size, void* d_ws, size_t ws_size,
                              hipStream_t stream) {
    (void)in_sizes; (void)n_in; (void)out_size; (void)ws_size;

    const float* x   = (const float*)d_in[0];   // (8,8192,1,64) fp32
    const float* emb = (const float*)d_in[1];   // (1,1024,64)   fp32

    float*    ee     = (float*)d_ws;                       // 1024 f32
    unsigned* counts = (unsigned*)((char*)d_ws + 4096);    // 1024 u32

    float* out  = (float*)d_out;
    float* out0 = out;                          // 65536*64
    float* out1 = out0 + (size_t)NS_TOT * VV;   // 65536
    float* out2 = out1 + NS_TOT;                // 65536
    float* ent  = out2 + NS_TOT;                // 1

    vq_prep<<<4, 256, 0, stream>>>(emb, ee, counts);
    vq_main<<<NS_TOT / 128, 256, 0, stream>>>(x, emb, ee, counts, out0, out1, out2);
    vq_entropy<<<1, 32, 0, stream>>>(counts, ent);
}